// SurrogateLIFCell_72610717106764
// MI455X (gfx1250) — compile-verified
//
#include <hip/hip_runtime.h>
#include <cstdint>

// LIF scan: v = v*TAU + I_t; spike = (v > V_TH); v -= spike*V_TH.
// T=64, N=4096, D=128. Memory-bound: 402 MB HBM traffic, ~17 us roofline
// at 23.3 TB/s. One thread per float4 column, scan over T in registers,
// input streamed through an 8-deep async global->LDS ring (ASYNCcnt),
// outputs written with non-temporal b128 stores.

typedef float v4f __attribute__((ext_vector_type(4)));

constexpr int T     = 64;
constexpr int N     = 4096;
constexpr int D     = 128;
constexpr int ND    = N * D;          // elements per time slice (524288)
constexpr int NDV   = ND / 4;         // float4 columns per slice  (131072)
constexpr int BLOCK = 256;            // 8 wave32 per block
constexpr int DEPTH = 8;              // time-steps in flight per wave

__global__ __launch_bounds__(BLOCK) void lif_scan_async(
    const float* __restrict__ I_in,   // [T, ND]
    float*       __restrict__ spikes, // [T, ND]
    float*       __restrict__ v_mem)  // [T, ND]
{
    __shared__ v4f smem[DEPTH][BLOCK];   // 32 KB ring buffer

    const int tid = threadIdx.x;
    const int col = blockIdx.x * BLOCK + tid;      // float4 column id

    // 64-bit global byte address of this thread's element at t=0.
    uint64_t gaddr = (uint64_t)(uintptr_t)(I_in + (size_t)col * 4);
    const uint64_t gstride = (uint64_t)NDV * sizeof(v4f);  // bytes per step

    // ---- prime the pipeline: DEPTH async loads in flight ----
#pragma unroll
    for (int p = 0; p < DEPTH; ++p) {
        uint32_t laddr = (uint32_t)(uintptr_t)&smem[p][tid];  // LDS byte offset
        asm volatile("global_load_async_to_lds_b128 %0, %1, off"
                     :: "v"(laddr), "v"(gaddr) : "memory");
        gaddr += gstride;
    }

    v4f v = {0.0f, 0.0f, 0.0f, 0.0f};
    const float tau = 0.95f;

    v4f* sp = (v4f*)spikes + col;
    v4f* vp = (v4f*)v_mem  + col;

    // ---- steady state: consume slot t, refill it for t+DEPTH ----
    for (int t = 0; t < T - DEPTH; ++t) {
        const int slot = t & (DEPTH - 1);

        // Async loads retire in order: <=DEPTH-1 in flight => oldest done.
        asm volatile("s_wait_asynccnt 0x7" ::: "memory");
        v4f I = smem[slot][tid];
        // DS and ASYNC ops are mutually unordered: drain the LDS read
        // before the async engine may overwrite this slot.
        asm volatile("s_wait_dscnt 0x0" ::: "memory");

        uint32_t laddr = (uint32_t)(uintptr_t)&smem[slot][tid];
        asm volatile("global_load_async_to_lds_b128 %0, %1, off"
                     :: "v"(laddr), "v"(gaddr) : "memory");
        gaddr += gstride;

        // LIF update
        v = v * tau + I;
        v4f s;
        s.x = v.x > 1.0f ? 1.0f : 0.0f;
        s.y = v.y > 1.0f ? 1.0f : 0.0f;
        s.z = v.z > 1.0f ? 1.0f : 0.0f;
        s.w = v.w > 1.0f ? 1.0f : 0.0f;
        v = v - s;                       // V_TH = 1.0

        __builtin_nontemporal_store(s, sp);
        __builtin_nontemporal_store(v, vp);
        sp += NDV;
        vp += NDV;
    }

    // ---- drain: the last DEPTH slices are already (or soon) in LDS ----
    asm volatile("s_wait_asynccnt 0x0" ::: "memory");
#pragma unroll
    for (int t = T - DEPTH; t < T; ++t) {
        v4f I = smem[t & (DEPTH - 1)][tid];

        v = v * tau + I;
        v4f s;
        s.x = v.x > 1.0f ? 1.0f : 0.0f;
        s.y = v.y > 1.0f ? 1.0f : 0.0f;
        s.z = v.z > 1.0f ? 1.0f : 0.0f;
        s.w = v.w > 1.0f ? 1.0f : 0.0f;
        v = v - s;

        __builtin_nontemporal_store(s, sp);
        __builtin_nontemporal_store(v, vp);
        sp += NDV;
        vp += NDV;
    }
}

extern "C" void kernel_launch(void* const* d_in, const int* in_sizes, int n_in,
                              void* d_out, int out_size, void* d_ws, size_t ws_size,
                              hipStream_t stream) {
    (void)in_sizes; (void)n_in; (void)out_size; (void)d_ws; (void)ws_size;

    const float* I_in = (const float*)d_in[0];
    float* out    = (float*)d_out;
    float* spikes = out;                       // first T*N*D floats
    float* vmem   = out + (size_t)T * ND;      // second T*N*D floats

    dim3 grid(NDV / BLOCK);                    // 512 blocks
    dim3 block(BLOCK);
    lif_scan_async<<<grid, block, 0, stream>>>(I_in, spikes, vmem);
}